// Quantizer_60481729462894
// MI455X (gfx1250) — compile-verified
//
#include <hip/hip_runtime.h>
#include <math.h>

// FP4-style block quantizer for MI455X (gfx1250, wave32).
// One 32-element quant block == 8 lanes x float4. Absmax reduction is a
// 3-step ds_swizzle XOR butterfly confined to 8-lane groups. All traffic is
// non-temporal B128 (streaming 512MB >> 192MB L2). Fast path is guardless so
// the 8 loads issue back-to-back (clause-able) with no EXEC juggling.

typedef float v4f __attribute__((ext_vector_type(4)));

#define TPB 256
#define UNROLL 8

// Max over the 8-lane group this lane belongs to (lanes sharing tid>>3).
// Operands are |x| (non-negative) so IEEE bit patterns order as ints.
// ds_swizzle group-of-32 encoding: offset[14:10]=xor, [9:5]=or, [4:0]=and.
__device__ __forceinline__ float blockmax8(float a) {
  int i = __float_as_int(a);
  i = max(i, __builtin_amdgcn_ds_swizzle(i, (1 << 10) | 0x1F));
  i = max(i, __builtin_amdgcn_ds_swizzle(i, (2 << 10) | 0x1F));
  i = max(i, __builtin_amdgcn_ds_swizzle(i, (4 << 10) | 0x1F));
  return __int_as_float(i);
}

// Snap non-negative scaled value to {0,.5,1,1.5,2,3,4,6}, half-up midpoints
// (0.25,0.75,1.25,1.75 | 2.5,3.5 | 5).
__device__ __forceinline__ float quant_abs(float a) {
  float r1 = floorf(fmaf(a, 2.0f, 0.5f)) * 0.5f;   // 0.5-step region, a < 2.25
  float r2 = fminf(floorf(a + 0.5f), 4.0f);        // {2,3,4} region
  float q  = (a < 2.25f) ? r1 : r2;
  return (a >= 5.0f) ? 6.0f : q;                   // top bucket
}

template <bool GUARD>
__global__ __launch_bounds__(TPB) void fp4_block_quant(
    const float* __restrict__ x, float* __restrict__ out, int n4) {
  const int tid  = threadIdx.x;
  const int base = blockIdx.x * (TPB * UNROLL) + tid;
  const v4f* __restrict__ xv = (const v4f*)x;
  v4f* __restrict__ ov       = (v4f*)out;

  v4f v[UNROLL];
#pragma unroll
  for (int k = 0; k < UNROLL; ++k) {
    const int i = base + k * TPB;
    if (GUARD) {
      v4f t = {0.0f, 0.0f, 0.0f, 0.0f};
      if (i < n4) t = __builtin_nontemporal_load(xv + i);
      v[k] = t;
    } else {
      v[k] = __builtin_nontemporal_load(xv + i);  // global_load_b128 TH_NT
    }
  }

#pragma unroll
  for (int k = 0; k < UNROLL; ++k) {
    const int i = base + k * TPB;

    float a0 = fabsf(v[k].x), a1 = fabsf(v[k].y);
    float a2 = fabsf(v[k].z), a3 = fabsf(v[k].w);
    float am = fmaxf(fmaxf(a0, a1), fmaxf(a2, a3));
    am = blockmax8(am);                 // absmax over the 32-element block
    // absmax==0 -> FLT_MIN; any absmax <= 2^-125 clamps to e=-127 below,
    // identical to the reference's exponent clamp, so this is exact.
    am = fmaxf(am, 1.17549435e-38f);

    // floor(log2(am)) - EHIGH via the exponent field (am is normal here).
    int e = (int)((__float_as_uint(am) >> 23) & 0xFFu) - 129;
    e = (e < -127) ? -127 : e;

    const float m    = ldexpf(0.75f, -e);               // 0.75/scale (exact pow2)
    const float omul = ldexpf(1.0f, e) * (4.0f / 3.0f); // scale * 4/3

    v4f r;
    r.x = copysignf(quant_abs(a0 * m), v[k].x) * omul;
    r.y = copysignf(quant_abs(a1 * m), v[k].y) * omul;
    r.z = copysignf(quant_abs(a2 * m), v[k].z) * omul;
    r.w = copysignf(quant_abs(a3 * m), v[k].w) * omul;

    if (GUARD) {
      if (i < n4) __builtin_nontemporal_store(r, ov + i);
    } else {
      __builtin_nontemporal_store(r, ov + i);           // global_store_b128 TH_NT
    }
  }
}

extern "C" void kernel_launch(void* const* d_in, const int* in_sizes, int n_in,
                              void* d_out, int out_size, void* d_ws, size_t ws_size,
                              hipStream_t stream) {
  (void)n_in; (void)out_size; (void)d_ws; (void)ws_size;
  const float* x = (const float*)d_in[0];
  float* out     = (float*)d_out;
  const int n  = in_sizes[0];       // 4*4096*4096 here; multiple of 32 by spec
  const int n4 = n >> 2;            // float4 count
  const int per_block = TPB * UNROLL;
  if (n4 % per_block == 0) {
    // Exact tiling (true for the harness shape): guardless streaming kernel.
    fp4_block_quant<false><<<n4 / per_block, TPB, 0, stream>>>(x, out, n4);
  } else {
    const int grid = (n4 + per_block - 1) / per_block;
    fp4_block_quant<true><<<grid, TPB, 0, stream>>>(x, out, n4);
  }
}